// DWT2_10075993276786
// MI455X (gfx1250) — compile-verified
//
#include <hip/hip_runtime.h>
#include <hip/hip_bf16.h>

typedef __attribute__((ext_vector_type(2))) float v2f;
typedef __attribute__((ext_vector_type(8))) float v8f;

// Haar element H[r][k] = 0.5 * (-1)^((r0&k1) ^ (r1&k0))
__device__ __forceinline__ float haar_elem(int r, int k) {
    int s = ((r & 1) & ((k >> 1) & 1)) ^ (((r >> 1) & 1) & (k & 1));
    return s ? -0.5f : 0.5f;
}

// x: [16,64,256,256] f32  ->  out: [16,256,128,128] f32 (out ch = r*64 + c)
// One wave handles 32 consecutive output columns j of one output row i of one
// (n,ch) plane, via two V_WMMA_F32_16X16X4_F32 ops.
__global__ __launch_bounds__(256) void haar_dwt2_wmma(const float* __restrict__ x,
                                                      float* __restrict__ out) {
    const int lane = threadIdx.x & 31;
    const unsigned wid = blockIdx.x * 8u + (threadIdx.x >> 5); // 524288 waves

    const int jchunk = wid & 3;           // 4 chunks of 32 columns (w2 = 128)
    const int i      = (wid >> 2) & 127;  // output row (h2 = 128)
    const int nc     = wid >> 9;          // n*64 + ch, 0..1023
    const int n      = nc >> 6;
    const int ch     = nc & 63;

    const int hpar  = lane >> 4;          // lanes 0-15 -> row 2i, 16-31 -> row 2i+1
    const int jl    = lane & 15;
    const int jbase = jchunk * 32;

    // ---- B matrices (4x16 f32; K rows = [x00,x01,x10,x11]) straight from memory.
    // Lane L   : VGPR pair = (B[0][L], B[1][L]) = (x00,x01) of position jbase+L
    // Lane L+16: VGPR pair = (B[2][L], B[3][L]) = (x10,x11) of the same position.
    const float* src = x + ((size_t)nc * 256 + (size_t)(2 * i + hpar)) * 256;
    v2f bA = __builtin_nontemporal_load((const v2f*)(src + 2 * (jbase + jl)));
    v2f bB = __builtin_nontemporal_load((const v2f*)(src + 2 * (jbase + 16 + jl)));

    // ---- Constant A (16x4): Haar rows in rows 0-3 and 8-11, zeros elsewhere.
    // Lane L holds (A[m][0],A[m][1]); lane L+16 holds (A[m][2],A[m][3]), m = lane&15.
    const int  m     = lane & 15;
    const int  r     = m & 3;
    const bool valid = ((m >> 2) & 1) == 0;   // rows 0-3 and 8-11
    v2f a;
    a.x = valid ? haar_elem(r, 2 * hpar)     : 0.0f;
    a.y = valid ? haar_elem(r, 2 * hpar + 1) : 0.0f;

    // ---- D = A x B (+0):  D vgpr v, lanes 0-15 = y_v of 16 positions (M=v),
    //                        lanes 16-31 = same y_v duplicated (M=v+8).
    v8f c = {};
    v8f dA = __builtin_amdgcn_wmma_f32_16x16x4_f32(
        false, a, false, bA, (short)0, c, false, false);
    v8f dB = __builtin_amdgcn_wmma_f32_16x16x4_f32(
        false, a, false, bB, (short)0, c, false, false);

    // Merge the two 16-position tiles into full 32-lane rows:
    // lanes 0-15 take dA (positions jbase+lane), lanes 16-31 take dB's
    // duplicated half (positions jbase+16+(lane-16) = jbase+lane).
    const bool lo = lane < 16;
    const float y0 = lo ? dA[0] : dB[0];
    const float y1 = lo ? dA[1] : dB[1];
    const float y2 = lo ? dA[2] : dB[2];
    const float y3 = lo ? dA[3] : dB[3];

    // out[n][r*64+ch][i][jbase+lane]; plane stride between r-components:
    size_t ob = (((size_t)n * 256 + ch) * 16384) + (size_t)i * 128 + jbase + lane;
    const size_t vs = (size_t)64 * 16384;
    __builtin_nontemporal_store(y0, out + ob);
    __builtin_nontemporal_store(y1, out + ob + vs);
    __builtin_nontemporal_store(y2, out + ob + 2 * vs);
    __builtin_nontemporal_store(y3, out + ob + 3 * vs);
}

extern "C" void kernel_launch(void* const* d_in, const int* in_sizes, int n_in,
                              void* d_out, int out_size, void* d_ws, size_t ws_size,
                              hipStream_t stream) {
    const float* x = (const float*)d_in[0];
    float* out = (float*)d_out;
    // 16*64*128*(128/32) = 524288 waves; 8 waves per 256-thread block.
    dim3 grid(65536), block(256);
    hipLaunchKernelGGL(haar_dwt2_wmma, grid, block, 0, stream, x, out);
    (void)in_sizes; (void)n_in; (void)out_size; (void)d_ws; (void)ws_size;
}